// CUDASpatialHash_24927990186056
// MI455X (gfx1250) — compile-verified
//
#include <hip/hip_runtime.h>
#include <stdint.h>

#define CELL 0.01f
#define HASH_SIZE 1000000
#define MAXN 64

typedef __attribute__((ext_vector_type(2))) float v2f;
typedef __attribute__((ext_vector_type(8))) float v8f;
typedef __attribute__((address_space(3))) int lds_int;

__device__ __forceinline__ int hash_cell(int cx, int cy, int cz) {
    // int32 wrapping multiply like the reference (jnp int32 semantics)
    int h = (int)((unsigned)cx * 73856093u +
                  (unsigned)cy * 19349663u +
                  (unsigned)cz * 83492791u);
    h = h < 0 ? -h : h;
    return h % HASH_SIZE;
}

// ---------------- Pass 1: build spatial hash table ----------------
__global__ void build_hash_kernel(const float* __restrict__ pos, int N,
                                  int* __restrict__ counts,
                                  int* __restrict__ table) {
    int i = blockIdx.x * blockDim.x + threadIdx.x;
    if (i >= N) return;
    float x = pos[3 * i + 0], y = pos[3 * i + 1], z = pos[3 * i + 2];
    int cx = (int)floorf(x / CELL);
    int cy = (int)floorf(y / CELL);
    int cz = (int)floorf(z / CELL);
    int h = hash_cell(cx, cy, cz);
    int slot = atomicAdd(&counts[h], 1);
    if (slot < MAXN) table[(size_t)h * MAXN + slot] = i;
}

// ---------------- Pass 2: query. One wave32 per query. ----------------
__launch_bounds__(256)
__global__ void query_kernel(const float* __restrict__ pos, int N,
                             const float* __restrict__ qry, int Q,
                             const float* __restrict__ radp,
                             const int* __restrict__ counts,
                             const int* __restrict__ table,
                             int* __restrict__ out_nb,
                             int* __restrict__ out_cnt) {
    // per-wave double buffer for one bucket row (64 ids = 256B), filled by
    // global_load_async_to_lds_b64 (ASYNCcnt-tracked DMA, no VGPR round-trip)
    __shared__ int s_buf[8][2][MAXN];

    const int lane = threadIdx.x & 31;   // wave32
    const int w    = threadIdx.x >> 5;
    const int qi   = blockIdx.x * 8 + w;
    if (qi >= Q) return;

    const float radius = radp[0];
    const float qx = qry[3 * qi + 0];
    const float qy = qry[3 * qi + 1];
    const float qz = qry[3 * qi + 2];

    int* __restrict__ nb_row = out_nb + (size_t)qi * MAXN;
    // init this query's neighbor row to -1 (reference fill value)
    nb_row[lane]      = -1;
    nb_row[lane + 32] = -1;

    const int R    = (int)ceilf(radius / CELL);   // == 2 for radius 0.02
    const int side = 2 * R + 1;
    const int C    = side * side * side;          // 125 cells
    // d^2 <= r^2  <=>  (-2q)·p + |p|^2 <= r^2 - |q|^2
    const float thresh = radius * radius - (qx * qx + qy * qy + qz * qz);

    // hash of neighbor cell c (replicates reference's float round-trip)
    auto cellh = [&](int c) -> int {
        int ox = c / (side * side) - R;
        int oy = (c / side) % side - R;
        int oz = c % side - R;
        float nx = (qx / CELL + (float)ox) * CELL;
        float ny = (qy / CELL + (float)oy) * CELL;
        float nz = (qz / CELL + (float)oz) * CELL;
        return hash_cell((int)floorf(nx / CELL),
                         (int)floorf(ny / CELL),
                         (int)floorf(nz / CELL));
    };

    // async DMA of one 256B bucket row into LDS: 32 lanes x 8B
    auto fetch_row = [&](int h, int buf) {
        lds_int* lp = (lds_int*)&s_buf[w][buf][0];
        unsigned lds_addr = (unsigned)(unsigned long long)lp + 8u * (unsigned)lane;
        unsigned long long gsrc =
            (unsigned long long)(uintptr_t)(table + (size_t)h * MAXN) + 8ull * (unsigned)lane;
        asm volatile("global_load_async_to_lds_b64 %0, %1, off"
                     :: "v"(lds_addr), "v"(gsrc) : "memory");
    };

    int total = 0;
    int h_next = cellh(0);
    fetch_row(h_next, 0);
    int cnt_next = counts[h_next];        // count prefetched alongside the row

    for (int c = 0; c < C; ++c) {
        const int pb = c & 1;
        int cnt = cnt_next;
        if (cnt > MAXN) cnt = MAXN;

        if (c + 1 < C) {
            h_next = cellh(c + 1);
            fetch_row(h_next, (c + 1) & 1);           // prefetch next row
            cnt_next = counts[h_next];                // prefetch next count
            asm volatile("s_wait_asynccnt 0x1" ::: "memory");  // current row landed
        } else {
            asm volatile("s_wait_asynccnt 0x0" ::: "memory");
        }

        for (int base = 0; base < cnt; base += 16) {
            // lanes l and l+16 both own candidate (l&15) of this batch
            const int m   = lane & 15;
            const int cid = (base + m < cnt) ? s_buf[w][pb][base + m] : -1;
            float px = 1e9f, py = 1e9f, pz = 1e9f;   // sentinel -> fails test
            if (cid >= 0 && cid < N) {
                px = pos[3 * cid + 0];
                py = pos[3 * cid + 1];
                pz = pos[3 * cid + 2];
            }
            // A 16x4 f32 layout: lanes 0-15 hold K=0,1; lanes 16-31 hold K=2,3
            // A row = (px, py, pz, |p|^2) ; B col = (-2qx, -2qy, -2qz, 1)
            const bool hi = lane >= 16;
            v2f A, B;
            A.x = hi ? pz : px;
            A.y = hi ? (px * px + py * py + pz * pz) : py;
            B.x = hi ? (-2.0f * qz) : (-2.0f * qx);
            B.y = hi ? 1.0f : (-2.0f * qy);
            v8f Cacc = {};
            v8f D = __builtin_amdgcn_wmma_f32_16x16x4_f32(
                false, A, false, B, (short)0, Cacc, false, false);

            // D VGPR r: lanes 0-15 = row r, lanes 16-31 = row r+8; all columns
            // identical (B columns replicated), so ballot bit0/bit16 are the
            // pass flags for candidates base+r / base+r+8.
            unsigned pass = 0;
            #pragma unroll
            for (int r = 0; r < 8; ++r) {
                unsigned long long b = __ballot(D[r] <= thresh);
                pass |= ((unsigned)(b & 1ull)) << r;
                pass |= ((unsigned)((b >> 16) & 1ull)) << (r + 8);
            }
            const int rem = cnt - base;
            const unsigned valid = (rem >= 16) ? 0xFFFFu : ((1u << rem) - 1u);
            const unsigned hits  = pass & valid;      // wave-uniform hit mask

            // Ballot-rank parallel compaction: lane l (<16) owns candidate
            // base+l (its cid is already in a register). Output slot =
            // total + popcount of lower hit bits — preserves slot order.
            const bool mine = (lane < 16) && ((hits >> lane) & 1u);
            if (mine) {
                int slot_out = total + __popc(hits & ((1u << lane) - 1u));
                if (slot_out < MAXN) nb_row[slot_out] = cid;
            }
            total += __popc(hits);                    // uniform running count
        }
    }
    if (lane == 0) out_cnt[qi] = total > MAXN ? MAXN : total;
}

extern "C" void kernel_launch(void* const* d_in, const int* in_sizes, int n_in,
                              void* d_out, int out_size, void* d_ws, size_t ws_size,
                              hipStream_t stream) {
    const float* pos = (const float*)d_in[0];
    const float* qry = (const float*)d_in[1];
    const float* rad = (const float*)d_in[2];
    const int N = in_sizes[0] / 3;
    const int Q = in_sizes[1] / 3;

    int* counts = (int*)d_ws;                                    // 4 MB
    int* table  = (int*)((char*)d_ws + (size_t)HASH_SIZE * 4);   // 256 MB

    hipMemsetAsync(counts, 0, (size_t)HASH_SIZE * sizeof(int), stream);
    build_hash_kernel<<<(N + 255) / 256, 256, 0, stream>>>(pos, N, counts, table);

    int* out_nb  = (int*)d_out;
    int* out_cnt = out_nb + (size_t)Q * MAXN;
    query_kernel<<<(Q + 7) / 8, 256, 0, stream>>>(pos, N, qry, Q, rad,
                                                  counts, table, out_nb, out_cnt);
}